// SGATLayer_81870666596760
// MI455X (gfx1250) — compile-verified
//
#include <hip/hip_runtime.h>
#include <hip/hip_bf16.h>
#include <math.h>

// ---------------- problem constants (match reference) ----------------
#define NSC 2
#define Nn  25000
#define Ee  400000
#define E2e 800000
#define INF_ 128
#define EINF 64
#define Hh  4
#define Dd  32
#define HD  128

typedef __attribute__((ext_vector_type(2))) float v2f;
typedef __attribute__((ext_vector_type(8))) float v8f;

// ---------- monotonic float<->uint mapping for atomic float max ----------
__device__ __forceinline__ unsigned f2mono(float f) {
    unsigned u = __float_as_uint(f);
    return (u & 0x80000000u) ? ~u : (u | 0x80000000u);
}
__device__ __forceinline__ float mono2f(unsigned u) {
    unsigned b = (u & 0x80000000u) ? (u & 0x7fffffffu) : ~u;
    return __uint_as_float(b);
}

__device__ __forceinline__ float wave_sum(float v) {
    #pragma unroll
    for (int off = 16; off > 0; off >>= 1) v += __shfl_xor(v, off, 32);
    return v;
}

// ================= WMMA fp32 GEMM: out[M,128] = X[M,K] @ W[K,128] =================
// One block = 128 threads = 4 waves; block covers 16 rows x 128 cols.
// Wave w computes two 16x16 tiles at columns [32w, 32w+16), [32w+16, 32w+32).
// ISA layouts (CDNA5 ISA):
//   A 16x4 f32 : lane(l<16) holds M=l,   K=k,k+1 ; lane(l>=16) holds M=l-16, K=k+2,k+3
//   B 4x16 f32 : VGPR j, half h : row K = k + 2h + j, col N = lane&15
//   C/D 16x16  : VGPR i, half h : row M = i + 8h, col N = lane&15
template <int K>
__global__ __launch_bounds__(128) void gemm_wmma_f32(const float* __restrict__ X,
                                                     const float* __restrict__ W,
                                                     float* __restrict__ out, int M) {
    __shared__ float Wlds[K * 128];
    const int tid = threadIdx.x;
    for (int i = tid; i < K * 128; i += 128) Wlds[i] = W[i];
    __syncthreads();

    const int row0 = blockIdx.x * 16;
    const int wave = tid >> 5;
    const int lane = tid & 31;
    const int half = lane >> 4;
    const int m    = lane & 15;
    const int col0 = wave * 32;

    int r = row0 + m;
    if (r >= M) r = M - 1;             // clamp so EXEC stays all-1 through WMMA

    v8f c0 = {}, c1 = {};
    for (int k = 0; k < K; k += 4) {
        const int kk = k + 2 * half;
        v2f a, b0, b1;
        a.x  = X[(size_t)r * K + kk];
        a.y  = X[(size_t)r * K + kk + 1];
        b0.x = Wlds[kk * 128 + col0 + m];
        b0.y = Wlds[(kk + 1) * 128 + col0 + m];
        b1.x = Wlds[kk * 128 + col0 + 16 + m];
        b1.y = Wlds[(kk + 1) * 128 + col0 + 16 + m];
        c0 = __builtin_amdgcn_wmma_f32_16x16x4_f32(false, a, false, b0, (short)0, c0, false, false);
        c1 = __builtin_amdgcn_wmma_f32_16x16x4_f32(false, a, false, b1, (short)0, c1, false, false);
    }

    #pragma unroll
    for (int i = 0; i < 8; ++i) {
        const int rr = row0 + i + 8 * half;
        if (rr < M) {
            out[(size_t)rr * 128 + col0 + m]      = c0[i];
            out[(size_t)rr * 128 + col0 + 16 + m] = c1[i];
        }
    }
}

// ============ fold attention vectors into weights: out[k,h] = sum_d W[k,h*32+d]*av[h,d] ============
__global__ void fold_attn(const float* __restrict__ W, const float* __restrict__ av,
                          float* __restrict__ out, int K) {
    int k = blockIdx.x * blockDim.x + threadIdx.x;
    if (k >= K) return;
    #pragma unroll
    for (int h = 0; h < Hh; ++h) {
        float s = 0.f;
        #pragma unroll
        for (int d = 0; d < Dd; ++d) s += W[(size_t)k * HD + h * Dd + d] * av[h * Dd + d];
        out[k * Hh + h] = s;
    }
}

// ============ node scores: el[n,h] = X[n,:]·p[:,h], er likewise (wave per node) ============
__global__ void node_scores(const float* __restrict__ X, const float* __restrict__ p,
                            const float* __restrict__ r, float* __restrict__ el,
                            float* __restrict__ er, int NN, int K) {
    const int n    = (blockIdx.x * blockDim.x + threadIdx.x) >> 5;
    const int lane = threadIdx.x & 31;
    if (n >= NN) return;
    float aL[Hh] = {0, 0, 0, 0}, aR[Hh] = {0, 0, 0, 0};
    for (int k = lane; k < K; k += 32) {
        const float x = X[(size_t)n * K + k];
        #pragma unroll
        for (int h = 0; h < Hh; ++h) {
            aL[h] += x * p[k * Hh + h];
            aR[h] += x * r[k * Hh + h];
        }
    }
    #pragma unroll
    for (int h = 0; h < Hh; ++h) {
        const float sL = wave_sum(aL[h]);
        const float sR = wave_sum(aR[h]);
        if (lane == 0) { el[n * Hh + h] = sL; er[n * Hh + h] = sR; }
    }
}

// ============ pass 1: logits + atomic segment max (wave per edge) ============
__global__ void edge_logits(const int* __restrict__ src, const int* __restrict__ dst,
                            const float* __restrict__ ef, const float* __restrict__ q,
                            const float* __restrict__ el, const float* __restrict__ er,
                            float* __restrict__ logits, unsigned* __restrict__ maxmono,
                            int NE, int K) {
    const int e    = (blockIdx.x * blockDim.x + threadIdx.x) >> 5;
    const int lane = threadIdx.x & 31;
    if (e >= NE) return;
    float ee[Hh] = {0, 0, 0, 0};
    for (int k = lane; k < K; k += 32) {
        const float x = ef[(size_t)e * K + k];
        #pragma unroll
        for (int h = 0; h < Hh; ++h) ee[h] += x * q[k * Hh + h];
    }
    #pragma unroll
    for (int h = 0; h < Hh; ++h) ee[h] = wave_sum(ee[h]);
    if (lane == 0) {
        const int s = src[e], d = dst[e];
        #pragma unroll
        for (int h = 0; h < Hh; ++h) {
            float v = el[s * Hh + h] + er[d * Hh + h] + ee[h];
            v = (v > 0.f) ? v : 0.2f * v;               // leaky_relu(0.2)
            logits[(size_t)e * Hh + h] = v;
            atomicMax(&maxmono[d * Hh + h], f2mono(v));
        }
    }
}

// ============ pass 2: ex = exp(logit - m[dst]) (in place) + atomic denom ============
__global__ void edge_exp(const int* __restrict__ dst, const unsigned* __restrict__ maxmono,
                         float* __restrict__ logits_ex, float* __restrict__ denom, int NE) {
    const int e = blockIdx.x * blockDim.x + threadIdx.x;
    if (e >= NE) return;
    const int d = dst[e];
    #pragma unroll
    for (int h = 0; h < Hh; ++h) {
        const unsigned mu = maxmono[d * Hh + h];
        const float m = (mu == 0u) ? 0.f : mono2f(mu);   // untouched segment -> m = 0 (ref isfinite rule)
        const float ex = expf(logits_ex[(size_t)e * Hh + h] - m);
        logits_ex[(size_t)e * Hh + h] = ex;
        atomicAdd(&denom[d * Hh + h], ex);
    }
}

// ============ pass 3: rst[dst] += (ex/denom[dst]) * feat[src]  (wave per edge) ============
__global__ void aggregate(const int* __restrict__ src, const int* __restrict__ dst,
                          const float* __restrict__ ex, const float* __restrict__ denom,
                          const float* __restrict__ feat, float* __restrict__ rst, int NE) {
    const int e    = (blockIdx.x * blockDim.x + threadIdx.x) >> 5;
    const int lane = threadIdx.x & 31;
    if (e >= NE) return;
    const int s = src[e], d = dst[e];
    const int idx  = lane * 4;          // 4 consecutive dims per lane, all in one head
    const int head = idx >> 5;
    const float a = ex[(size_t)e * Hh + head] / denom[(size_t)d * Hh + head];
    const float4 f = *(const float4*)&feat[(size_t)s * HD + idx];
    float* rp = &rst[(size_t)d * HD + idx];
    atomicAdd(rp + 0, a * f.x);
    atomicAdd(rp + 1, a * f.y);
    atomicAdd(rp + 2, a * f.z);
    atomicAdd(rp + 3, a * f.w);
}

// ============ pass 4: elu ============
__global__ void elu_kernel(const float* __restrict__ in, float* __restrict__ out, size_t n) {
    const size_t i = (size_t)blockIdx.x * blockDim.x + threadIdx.x;
    if (i >= n) return;
    const float x = in[i];
    out[i] = (x > 0.f) ? x : expm1f(x);
}

// =====================================================================
extern "C" void kernel_launch(void* const* d_in, const int* in_sizes, int n_in,
                              void* d_out, int out_size, void* d_ws, size_t ws_size,
                              hipStream_t stream) {
    // inputs (complex 0 = first slice of each NSC-leading array)
    const int*   src  = (const int*)d_in[0];
    const int*   dst  = (const int*)d_in[1];
    const int*   esrc = (const int*)d_in[2];
    const int*   edst = (const int*)d_in[3];
    const float* h    = (const float*)d_in[4];
    const float* ef   = (const float*)d_in[5];   // [E, 64]
    const float* efu  = (const float*)d_in[6];   // [E2, 128]
    const float* Wn_g = (const float*)d_in[7];
    const float* We_g = (const float*)d_in[8];
    const float* al_g = (const float*)d_in[9];
    const float* ar_g = (const float*)d_in[10];
    const float* ae_g = (const float*)d_in[11];
    const float* Wn_u = (const float*)d_in[12];
    const float* We_u = (const float*)d_in[13];
    const float* al_u = (const float*)d_in[14];
    const float* ar_u = (const float*)d_in[15];
    const float* ae_u = (const float*)d_in[16];

    float* out_node = (float*)d_out;                       // [N,128]  (rst_g accumulator, elu'd in place)
    float* out_edge = (float*)d_out + (size_t)Nn * HD;     // [E,128]  (feat_u staging, then final)

    // ---- workspace layout ----
    float* ws = (float*)d_ws;
    size_t o = 0;
    float* feat_g   = ws + o; o += (size_t)Nn * HD;
    float* p_l      = ws + o; o += INF_ * Hh;
    float* r_l      = ws + o; o += INF_ * Hh;
    float* q_l      = ws + o; o += EINF * Hh;
    float* p_u      = ws + o; o += EINF * Hh;
    float* r_u      = ws + o; o += EINF * Hh;
    float* q_u      = ws + o; o += INF_ * Hh;
    float* el_g     = ws + o; o += (size_t)Nn * Hh;
    float* er_g     = ws + o; o += (size_t)Nn * Hh;
    float* lex_g    = ws + o; o += (size_t)Ee * Hh;        // logits, then ex (in place)
    unsigned* mm_g  = (unsigned*)(ws + o); o += (size_t)Nn * Hh;
    float* den_g    = ws + o; o += (size_t)Nn * Hh;
    float* el_u     = ws + o; o += (size_t)Ee * Hh;
    float* er_u     = ws + o; o += (size_t)Ee * Hh;
    float* lex_u    = ws + o; o += (size_t)E2e * Hh;
    unsigned* mm_u  = (unsigned*)(ws + o); o += (size_t)Ee * Hh;
    float* den_u    = ws + o; o += (size_t)Ee * Hh;
    float* rst_u    = ws + o; o += (size_t)Ee * HD;

    // ---- zero accumulators (0 == mono(-inf) for max buffers) ----
    hipMemsetAsync(out_node, 0, (size_t)Nn * HD * sizeof(float), stream);
    hipMemsetAsync(mm_g, 0, (size_t)Nn * Hh * sizeof(unsigned), stream);
    hipMemsetAsync(den_g, 0, (size_t)Nn * Hh * sizeof(float), stream);
    hipMemsetAsync(mm_u, 0, (size_t)Ee * Hh * sizeof(unsigned), stream);
    hipMemsetAsync(den_u, 0, (size_t)Ee * Hh * sizeof(float), stream);
    hipMemsetAsync(rst_u, 0, (size_t)Ee * HD * sizeof(float), stream);

    // ---- fold attention vectors into projection weights ----
    fold_attn<<<1, 128, 0, stream>>>(Wn_g, al_g, p_l, INF_);
    fold_attn<<<1, 128, 0, stream>>>(Wn_g, ar_g, r_l, INF_);
    fold_attn<<<1, 128, 0, stream>>>(We_g, ae_g, q_l, EINF);
    fold_attn<<<1, 128, 0, stream>>>(Wn_u, al_u, p_u, EINF);
    fold_attn<<<1, 128, 0, stream>>>(Wn_u, ar_u, r_u, EINF);
    fold_attn<<<1, 128, 0, stream>>>(We_u, ae_u, q_u, INF_);

    // ---- WMMA GEMMs: feat_g = h@Wn_g ; feat_u = ef@Wn_u (staged in out_edge) ----
    gemm_wmma_f32<INF_><<<(Nn + 15) / 16, 128, 0, stream>>>(h, Wn_g, feat_g, Nn);
    gemm_wmma_f32<EINF><<<(Ee + 15) / 16, 128, 0, stream>>>(ef, Wn_u, out_edge, Ee);

    // ---- node attention scores (folded) ----
    node_scores<<<(Nn * 32 + 255) / 256, 256, 0, stream>>>(h, p_l, r_l, el_g, er_g, Nn, INF_);
    node_scores<<<((size_t)Ee * 32 + 255) / 256, 256, 0, stream>>>(ef, p_u, r_u, el_u, er_u, Ee, EINF);

    // ---- edge softmax: logits + max, then exp + denom ----
    edge_logits<<<((size_t)Ee * 32 + 255) / 256, 256, 0, stream>>>(src, dst, ef, q_l, el_g, er_g,
                                                                   lex_g, mm_g, Ee, EINF);
    edge_logits<<<((size_t)E2e * 32 + 255) / 256, 256, 0, stream>>>(esrc, edst, efu, q_u, el_u, er_u,
                                                                    lex_u, mm_u, E2e, INF_);
    edge_exp<<<(Ee + 255) / 256, 256, 0, stream>>>(dst, mm_g, lex_g, den_g, Ee);
    edge_exp<<<(E2e + 255) / 256, 256, 0, stream>>>(edst, mm_u, lex_u, den_u, E2e);

    // ---- weighted aggregation ----
    aggregate<<<((size_t)Ee * 32 + 255) / 256, 256, 0, stream>>>(src, dst, lex_g, den_g,
                                                                 feat_g, out_node, Ee);
    aggregate<<<((size_t)E2e * 32 + 255) / 256, 256, 0, stream>>>(esrc, edst, lex_u, den_u,
                                                                  out_edge, rst_u, E2e);

    // ---- elu epilogues ----
    elu_kernel<<<(int)(((size_t)Nn * HD + 255) / 256), 256, 0, stream>>>(out_node, out_node, (size_t)Nn * HD);
    elu_kernel<<<(int)(((size_t)Ee * HD + 255) / 256), 256, 0, stream>>>(rst_u, out_edge, (size_t)Ee * HD);
}